// VectorQuantizer_46102178955959
// MI455X (gfx1250) — compile-verified
//
#include <hip/hip_runtime.h>

typedef __attribute__((ext_vector_type(2))) float v2f;
typedef __attribute__((ext_vector_type(4))) float v4f;
typedef __attribute__((ext_vector_type(8))) float v8f;
typedef __attribute__((ext_vector_type(4))) unsigned int u32x4;
typedef __attribute__((ext_vector_type(8))) int i32x8;
typedef __attribute__((ext_vector_type(4))) int i32x4;

#define KCB   1024
#define DEMB  64
#define BETA_ 0.25f
#define LDS_STRIDE 68   // 64 data floats + 4 pad floats (TDM pad) -> conflict-free

// ---------------------------------------------------------------------------
// Kernel 1: e_sq[k] = ||codebook[k]||^2 ; zero histogram + sum accumulator
// ---------------------------------------------------------------------------
__global__ void vq_prep(const float* __restrict__ cb,
                        float* __restrict__ esq,
                        float* __restrict__ counts,
                        float* __restrict__ sums) {
  int i = blockIdx.x * blockDim.x + threadIdx.x;
  if (i < KCB) {
    const v4f* row = (const v4f*)(cb + (size_t)i * DEMB);
    float s = 0.f;
#pragma unroll
    for (int j = 0; j < DEMB / 4; ++j) {
      v4f v = row[j];
      s += v.x * v.x + v.y * v.y + v.z * v.z + v.w * v.w;
    }
    esq[i]    = s;
    counts[i] = 0.f;
  }
  if (i == 0) sums[0] = 0.f;
}

// ---------------------------------------------------------------------------
// TDM: DMA one 16x64 f32 codebook tile into LDS with 4-DWORD row padding.
// D# packing per CDNA5 ISA.
// ---------------------------------------------------------------------------
__device__ __forceinline__ void tdm_load_tile(const float* gsrc,
                                              unsigned lds_off) {
  unsigned long long ga = (unsigned long long)(uintptr_t)gsrc;
  u32x4 g0;
  g0[0] = 1u;                                   // count=1, user desc
  g0[1] = lds_off;                              // lds_addr [63:32]
  g0[2] = (unsigned)(ga & 0xffffffffu);         // global_addr [95:64]
  g0[3] = (unsigned)((ga >> 32) & 0x1ffffffu)   // global_addr [120:96]
          | (2u << 30);                         // type=2 (image)
  i32x8 g1;
  g1[0] = (int)((2u << 16)      // data_size = 4 bytes
                | (1u << 20)    // pad_enable
                | (5u << 22)    // pad_interval: after every 64 DWORDs (1 row)
                | (3u << 25));  // pad_amount: 4 DWORDs
  g1[1] = (int)(64u << 16);     // tensor_dim0 = 64 (bits[79:48], low half)
  g1[2] = (int)(16u << 16);     // tensor_dim1 = 16 (bits[111:80], low half)
  g1[3] = (int)(64u << 16);     // tile_dim0 = 64  (bits[127:112])
  g1[4] = 16;                   // tile_dim1 = 16 ; tile_dim2 = 0
  g1[5] = 64;                   // tensor_dim0_stride = 64
  g1[6] = 0;
  g1[7] = 0;
  i32x4 z4 = {0, 0, 0, 0};
  i32x8 z8 = {0, 0, 0, 0, 0, 0, 0, 0};
  __builtin_amdgcn_tensor_load_to_lds(g0, g1, z4, z4, z8, 0);
}

// ---------------------------------------------------------------------------
// Kernel 2: per-wave 16-row tile; B tiles TDM-staged in LDS (double-buffered,
// shared by all 8 waves); B pre-loaded to VGPRs so the dual WMMA accumulation
// chains issue back-to-back; running argmin of (e_sq[col] - 2*dot); then
// gather z_q, write outputs, MSE + histogram.
// ---------------------------------------------------------------------------
__global__ void __launch_bounds__(256) vq_main(
    const float* __restrict__ z, const float* __restrict__ cb,
    const float* __restrict__ esq, float* __restrict__ zq_out,
    float* __restrict__ idx_out, float* __restrict__ counts,
    float* __restrict__ sums) {
  __shared__ float lds_b[2][16 * LDS_STRIDE];

  const int lane = threadIdx.x & 31;
  const int wave = threadIdx.x >> 5;
  const int r0   = (blockIdx.x * 8 + wave) * 16;   // first z-row of this wave
  const int m    = lane & 15;                      // row-in-tile / col-in-tile
  const int hsel = (lane >> 4) * 2;                // lanes 16-31: dims +2

  // ---- load A tile: rows r0..r0+15, ISA 16x4 f32 A layout, 16 k-chunks ----
  v2f a[16];
  {
    const float* zrow = z + (size_t)(r0 + m) * DEMB + hsel;
#pragma unroll
    for (int kk = 0; kk < 16; ++kk)
      a[kk] = *(const v2f*)(zrow + kk * 4);
  }

  float bestv[8];
  int   besti[8];
#pragma unroll
  for (int i = 0; i < 8; ++i) { bestv[i] = 3.4e38f; besti[i] = 0; }

  const unsigned lds_base = (unsigned)(uintptr_t)&lds_b[0][0];
  const unsigned buf_bytes = 16 * LDS_STRIDE * 4;

  // prologue: wave 0 DMAs tile 0 into buffer 0
  if (wave == 0) tdm_load_tile(cb, lds_base);

  // ---- sweep 64 column tiles of 16 codebook entries each ----
  for (int ct = 0; ct < KCB / 16; ++ct) {
    const int buf = ct & 1;
    if (wave == 0) __builtin_amdgcn_s_wait_tensorcnt(0);
    __syncthreads();   // tile ct resident; everyone done reading buf^1 (ct-1)
    if (wave == 0 && ct + 1 < KCB / 16)
      tdm_load_tile(cb + (size_t)(ct + 1) * 16 * DEMB,
                    lds_base + (unsigned)(buf ^ 1) * buf_bytes);

    const int col = ct * 16 + m;
    const float* bp = &lds_b[buf][m * LDS_STRIDE + hsel];

    // pre-load all 16 B pairs (clauses the ds_load_2addr_b64s, single wait)
    v2f bb[16];
#pragma unroll
    for (int kk = 0; kk < 16; ++kk)
      bb[kk] = *(const v2f*)(bp + kk * 4);

    v8f acc0 = {}, acc1 = {};
#pragma unroll
    for (int kk = 0; kk < 16; kk += 2) {
      acc0 = __builtin_amdgcn_wmma_f32_16x16x4_f32(false, a[kk], false,
                                                   bb[kk], (short)0, acc0,
                                                   false, false);
      acc1 = __builtin_amdgcn_wmma_f32_16x16x4_f32(false, a[kk + 1], false,
                                                   bb[kk + 1], (short)0, acc1,
                                                   false, false);
    }
    const float es = esq[col];
#pragma unroll
    for (int i = 0; i < 8; ++i) {
      float sc = es - 2.0f * acc1[i];
      sc = sc - 2.0f * acc0[i];        // row = i (lanes 0-15) / i+8 (16-31)
      if (sc < bestv[i]) { bestv[i] = sc; besti[i] = col; }
    }
  }

  // ---- argmin reduce across the 16 lanes of each half (cols live on lanes) -
#pragma unroll
  for (int off = 8; off >= 1; off >>= 1) {
#pragma unroll
    for (int i = 0; i < 8; ++i) {
      float ov = __shfl_xor(bestv[i], off, 32);
      int   oi = __shfl_xor(besti[i], off, 32);
      if (ov < bestv[i] || (ov == bestv[i] && oi < besti[i])) {
        bestv[i] = ov; besti[i] = oi;
      }
    }
  }
  // lane 0 -> rows r0+0..7 (regs 0..7); lane 16 -> rows r0+8..15.

  // ---- writeback: indices, histogram, z_q gather, squared-error partial ----
  float lsum = 0.f;
#pragma unroll
  for (int i = 0; i < 8; ++i) {
    const int idx_lo = __shfl(besti[i], 0, 32);
    const int idx_hi = __shfl(besti[i], 16, 32);
#pragma unroll
    for (int h = 0; h < 2; ++h) {
      const int row = r0 + i + h * 8;
      const int idx = h ? idx_hi : idx_lo;
      if (lane == 0) {
        idx_out[row] = (float)idx;
        atomicAdd(&counts[idx], 1.0f);
      }
      // 32 lanes x 2 floats cover all 64 dims of this row
      v2f q  = *(const v2f*)(cb + (size_t)idx * DEMB + lane * 2);
      v2f zz = *(const v2f*)(z + (size_t)row * DEMB + lane * 2);
      *(v2f*)(zq_out + (size_t)row * DEMB + lane * 2) = q;  // z_q_st == z_q
      float dx = zz.x - q.x, dy = zz.y - q.y;
      lsum += dx * dx + dy * dy;
    }
  }
#pragma unroll
  for (int off = 16; off >= 1; off >>= 1) lsum += __shfl_xor(lsum, off, 32);
  if (lane == 0) atomicAdd(&sums[0], lsum);
}

// ---------------------------------------------------------------------------
// Kernel 3: loss_vq = (1+beta)*mse ; perplexity = exp(-sum p log p)
// ---------------------------------------------------------------------------
__global__ void vq_final(const float* __restrict__ counts,
                         const float* __restrict__ sums,
                         float* __restrict__ out_loss,
                         float* __restrict__ out_ppl, int n) {
  __shared__ float sh[256];
  const float inv_n = 1.0f / (float)n;
  float hacc = 0.f;
  for (int i = threadIdx.x; i < KCB; i += 256) {
    float p = counts[i] * inv_n;
    hacc += p * logf(p + 1e-12f);
  }
  sh[threadIdx.x] = hacc;
  __syncthreads();
  for (int s = 128; s > 0; s >>= 1) {
    if (threadIdx.x < (unsigned)s) sh[threadIdx.x] += sh[threadIdx.x + s];
    __syncthreads();
  }
  if (threadIdx.x == 0) {
    *out_ppl  = expf(-sh[0]);
    *out_loss = (1.0f + BETA_) * sums[0] / ((float)n * (float)DEMB);
  }
}

// ---------------------------------------------------------------------------
extern "C" void kernel_launch(void* const* d_in, const int* in_sizes, int n_in,
                              void* d_out, int out_size, void* d_ws,
                              size_t ws_size, hipStream_t stream) {
  const float* z  = (const float*)d_in[0];   // [N, 64] flattened z_e
  const float* cb = (const float*)d_in[1];   // [1024, 64]
  const int n = in_sizes[0] / DEMB;          // 131072 rows

  float* ws     = (float*)d_ws;
  float* esq    = ws;              // [1024]
  float* counts = ws + KCB;        // [1024]
  float* sums   = ws + 2 * KCB;    // [1]

  float* out  = (float*)d_out;
  float* zq   = out;                               // N*D floats
  float* idxf = out + (size_t)n * DEMB;            // N floats
  float* loss = idxf + n;                          // 1
  float* ppl  = loss + 1;                          // 1

  vq_prep<<<dim3((KCB + 255) / 256), dim3(256), 0, stream>>>(cb, esq, counts,
                                                             sums);
  vq_main<<<dim3(n / 128), dim3(256), 0, stream>>>(z, cb, esq, zq, idxf,
                                                   counts, sums);
  vq_final<<<dim3(1), dim3(256), 0, stream>>>(counts, sums, loss, ppl, n);
}